// GCN_14207751815505
// MI455X (gfx1250) — compile-verified
//
#include <hip/hip_runtime.h>

#define NN   100000
#define INC  256
#define HIDC 128
#define NE   1600000
#define NPOS 200000
#define NNEG 200000

typedef __attribute__((ext_vector_type(2))) float v2f;
typedef __attribute__((ext_vector_type(8))) float v8f;

// ---------------- degree / norm ----------------
__global__ void k_init_dinv(float* dinv) {
  int i = blockIdx.x * blockDim.x + threadIdx.x;
  if (i < NN) dinv[i] = 1.0f;              // self-loop contributes 1 to degree
}

__global__ void k_count_deg(float* dinv, const int* __restrict__ col) {
  int i = blockIdx.x * blockDim.x + threadIdx.x;
  if (i < NE) unsafeAtomicAdd(&dinv[col[i]], 1.0f);
}

__global__ void k_finish_dinv(float* dinv) {
  int i = blockIdx.x * blockDim.x + threadIdx.x;
  if (i < NN) dinv[i] = 1.0f / sqrtf(dinv[i]);
}

// ---------------- W transpose (for contiguous B fragments) ----------------
__global__ void k_transpose(const float* __restrict__ W, float* __restrict__ WT,
                            int K, int N) {
  int i = blockIdx.x * blockDim.x + threadIdx.x;
  if (i >= K * N) return;
  int k = i / N, n = i - k * N;
  WT[n * K + k] = W[i];
}

// ---------------- fp32 WMMA GEMM: C[NN x 128] = A[NN x K] * B, B as WT[128 x K]
// NN % 16 == 0, so no M-tail: no divergence anywhere, loop fully pipelines.
// One wave per 16-row strip; 8 WMMA tiles span the 128-wide output.
template <int K>
__global__ __launch_bounds__(256) void k_gemm_wmma(const float* __restrict__ A,
                                                   const float* __restrict__ WT,
                                                   float* __restrict__ C) {
  int wave = (blockIdx.x * blockDim.x + threadIdx.x) >> 5;
  int lane = threadIdx.x & 31;
  int m0 = wave * 16;
  if (m0 >= NN) return;                    // wave-uniform (grid tail only)
  int half = lane >> 4;                    // 0 -> K sub 0..1, 1 -> K sub 2..3
  int l15  = lane & 15;
  const float* aptr  = A  + (size_t)(m0 + l15) * K + half * 2;
  const float* bbase = WT + (size_t)l15 * K + half * 2;

  v8f acc[8];
#pragma unroll
  for (int t = 0; t < 8; ++t) acc[t] = (v8f)(0.0f);

  __builtin_prefetch(aptr, 0, 0);          // global_prefetch_b8 on A stream
#pragma unroll 2
  for (int k = 0; k < K; k += 4) {
    v2f a = *(const v2f*)(aptr + k);
#pragma unroll
    for (int t = 0; t < 8; ++t) {
      v2f b = *(const v2f*)(bbase + (size_t)t * 16 * K + k);
      acc[t] = __builtin_amdgcn_wmma_f32_16x16x4_f32(
          false, a, false, b, (short)0, acc[t], false, false);
    }
  }

  // C/D layout: VGPR i -> M = m0 + i + 8*half, N = t*16 + l15
#pragma unroll
  for (int t = 0; t < 8; ++t) {
#pragma unroll
    for (int i = 0; i < 8; ++i) {
      C[(size_t)(m0 + i + half * 8) * HIDC + t * 16 + l15] = acc[t][i];
    }
  }
}

// ---------------- aggregation ----------------
__global__ void k_selfloop(float* __restrict__ out, const float* __restrict__ h,
                           const float* __restrict__ dinv) {
  int i = blockIdx.x * blockDim.x + threadIdx.x;     // over NN * 32 float4 groups
  if (i >= NN * (HIDC / 4)) return;
  int n = i >> 5;
  float d = dinv[n];
  float s = d * d;
  float4 v = ((const float4*)h)[i];
  ((float4*)out)[i] = make_float4(v.x * s, v.y * s, v.z * s, v.w * s);
}

__global__ void k_edge_agg(float* __restrict__ out, const float* __restrict__ h,
                           const float* __restrict__ dinv,
                           const int* __restrict__ row, const int* __restrict__ col) {
  long long t = (long long)blockIdx.x * blockDim.x + threadIdx.x;
  if (t >= (long long)NE * (HIDC / 4)) return;
  int e = (int)(t >> 5);
  int g = (int)(t & 31);
  int r = row[e], c = col[e];
  float norm = dinv[r] * dinv[c];
  float4 v = ((const float4*)(h + (size_t)r * HIDC))[g];
  float* dst = out + (size_t)c * HIDC + g * 4;
  unsafeAtomicAdd(dst + 0, v.x * norm);
  unsafeAtomicAdd(dst + 1, v.y * norm);
  unsafeAtomicAdd(dst + 2, v.z * norm);
  unsafeAtomicAdd(dst + 3, v.w * norm);
}

__global__ void k_bias_act(float* __restrict__ z, const float* __restrict__ b,
                           int do_relu) {
  int i = blockIdx.x * blockDim.x + threadIdx.x;
  if (i >= NN * HIDC) return;
  int f = i & (HIDC - 1);
  float v = z[i] + b[f];
  if (do_relu) v = fmaxf(v, 0.0f);
  z[i] = v;
}

// ---------------- decoder: one wave32 per edge pair ----------------
__global__ void k_decode(const float* __restrict__ z, const int* __restrict__ pos,
                         const int* __restrict__ neg, float* __restrict__ out) {
  int w = (blockIdx.x * blockDim.x + threadIdx.x) >> 5;
  int lane = threadIdx.x & 31;
  if (w >= NPOS + NNEG) return;            // wave-uniform
  int a, b;
  if (w < NPOS) { a = pos[w]; b = pos[NPOS + w]; }
  else          { int j = w - NPOS; a = neg[j]; b = neg[NNEG + j]; }
  float4 va = ((const float4*)(z + (size_t)a * HIDC))[lane];
  float4 vb = ((const float4*)(z + (size_t)b * HIDC))[lane];
  float s = va.x * vb.x + va.y * vb.y + va.z * vb.z + va.w * vb.w;
#pragma unroll
  for (int off = 16; off > 0; off >>= 1) s += __shfl_xor(s, off, 32);
  if (lane == 0) out[w] = s;
}

// ---------------- host ----------------
extern "C" void kernel_launch(void* const* d_in, const int* in_sizes, int n_in,
                              void* d_out, int out_size, void* d_ws, size_t ws_size,
                              hipStream_t stream) {
  const float* x  = (const float*)d_in[0];
  const int*   ei = (const int*)d_in[1];
  const int*   pos = (const int*)d_in[2];
  const int*   neg = (const int*)d_in[3];
  const float* W1 = (const float*)d_in[4];
  const float* b1 = (const float*)d_in[5];
  const float* W2 = (const float*)d_in[6];
  const float* b2 = (const float*)d_in[7];
  float* logits = (float*)d_out;

  char* ws = (char*)d_ws;
  float* dinv = (float*)ws;                               // 400 KB
  float* WT   = (float*)(ws + (512u << 10));              // 128 KB max
  float* bufH = (float*)(ws + (1u << 20));                // 51.2 MB
  float* bufZ = (float*)(ws + (1u << 20) + (52u << 20));  // 51.2 MB

  const int* row = ei;
  const int* col = ei + NE;
  const int B = 256;

  // symmetric normalization coefficients
  k_init_dinv<<<(NN + B - 1) / B, B, 0, stream>>>(dinv);
  k_count_deg<<<(NE + B - 1) / B, B, 0, stream>>>(dinv, col);
  k_finish_dinv<<<(NN + B - 1) / B, B, 0, stream>>>(dinv);

  int gemm_blocks = (((NN + 15) / 16) + 7) / 8;           // 8 waves per block
  long long agg_t = (long long)NE * 32;
  int agg_blocks = (int)((agg_t + B - 1) / B);
  int sl_blocks  = (NN * 32 + B - 1) / B;
  int ba_blocks  = (NN * HIDC + B - 1) / B;

  // layer 1: h = x @ W1 ; z = relu(aggregate(h) + b1)
  k_transpose<<<(INC * HIDC + B - 1) / B, B, 0, stream>>>(W1, WT, INC, HIDC);
  k_gemm_wmma<INC><<<gemm_blocks, B, 0, stream>>>(x, WT, bufH);
  k_selfloop<<<sl_blocks, B, 0, stream>>>(bufZ, bufH, dinv);
  k_edge_agg<<<agg_blocks, B, 0, stream>>>(bufZ, bufH, dinv, row, col);
  k_bias_act<<<ba_blocks, B, 0, stream>>>(bufZ, b1, 1);

  // layer 2: h = z @ W2 ; z2 = aggregate(h) + b2
  k_transpose<<<(HIDC * HIDC + B - 1) / B, B, 0, stream>>>(W2, WT, HIDC, HIDC);
  k_gemm_wmma<HIDC><<<gemm_blocks, B, 0, stream>>>(bufZ, WT, bufH);
  k_selfloop<<<sl_blocks, B, 0, stream>>>(bufZ, bufH, dinv);
  k_edge_agg<<<agg_blocks, B, 0, stream>>>(bufZ, bufH, dinv, row, col);
  k_bias_act<<<ba_blocks, B, 0, stream>>>(bufZ, b2, 0);

  // decoder
  int dec_blocks = ((NPOS + NNEG) * 32 + B - 1) / B;
  k_decode<<<dec_blocks, B, 0, stream>>>(bufZ, pos, neg, logits);
}